// TwoLayerGraphSAGE_35390530519865
// MI455X (gfx1250) — compile-verified
//
#include <hip/hip_runtime.h>

#define NNODES 100000
#define NEDGES 1600000
#define FDIM   128   // IN_F == H1 == 128 (aggregation width for both layers)

typedef __attribute__((ext_vector_type(2))) float v2f;
typedef __attribute__((ext_vector_type(8))) float v8f;

// ---------------------------------------------------------------- zero scratch
__global__ void zero_f4_kernel(float4* __restrict__ p, int n4) {
    int i = blockIdx.x * blockDim.x + threadIdx.x;
    int stride = gridDim.x * blockDim.x;
    float4 z = {0.f, 0.f, 0.f, 0.f};
    for (; i < n4; i += stride) p[i] = z;
}

// ---------------------------------------------------------------- in-degree
__global__ void deg_count_kernel(const int* __restrict__ dst,
                                 float* __restrict__ deg, int nEdges) {
    int e = blockIdx.x * blockDim.x + threadIdx.x;
    if (e < nEdges) unsafeAtomicAdd(&deg[dst[e]], 1.0f);   // global_atomic_add_f32
}

// ------------------------------------------------- edge scatter: one wave/edge
// lane l moves 4 consecutive floats -> wave32 covers the whole 128-float row.
__global__ void scatter_add_kernel(const float* __restrict__ X,
                                   const int* __restrict__ src,
                                   const int* __restrict__ dst,
                                   float* __restrict__ agg, int nEdges) {
    int wave = (blockIdx.x * blockDim.x + threadIdx.x) >> 5;
    int lane = threadIdx.x & 31;
    if (wave >= nEdges) return;

    int s = src[wave];
    int d = dst[wave];

    // prefetch a future edge's source row toward L2 (global_prefetch_b8)
    int ef = wave + 4096;
    if (ef < nEdges) {
        __builtin_prefetch(X + (size_t)src[ef] * FDIM + lane * 4, 0, 1);
    }

    const float4 v = *(const float4*)(X + (size_t)s * FDIM + lane * 4);
    float* p = agg + (size_t)d * FDIM + lane * 4;
    unsafeAtomicAdd(p + 0, v.x);
    unsafeAtomicAdd(p + 1, v.y);
    unsafeAtomicAdd(p + 2, v.z);
    unsafeAtomicAdd(p + 3, v.w);
}

// ------------------------------------------------- fused SAGE layer via WMMA
// out[r, c] = act( X[r,:] @ Wself[:,c] + (AGG[r,:]/max(deg[r],1)) @ Wneigh[:,c] + b[c] )
// One wave32 computes one 16x16 tile using V_WMMA_F32_16X16X4_F32.
// A frag (16x4 f32): lanes 0-15 row M=l hold K=k,k+1; lanes 16-31 hold K=k+2,k+3.
// B frag (4x16 f32): VGPR0 = row k (lanes 0-15) / row k+2 (lanes 16-31); VGPR1 = k+1 / k+3.
__global__ void sage_gemm_kernel(const float* __restrict__ X,
                                 const float* __restrict__ AGG,
                                 const float* __restrict__ deg,
                                 const float* __restrict__ Wself,
                                 const float* __restrict__ Wneigh,
                                 const float* __restrict__ bias,
                                 float* __restrict__ out,
                                 int K, int Fout, int ntilesN, int do_relu) {
    const int lane  = threadIdx.x & 31;
    const int wib   = threadIdx.x >> 5;
    const int gwave = blockIdx.x * (blockDim.x >> 5) + wib;
    const int tileN = gwave % ntilesN;
    const int tileM = gwave / ntilesN;
    const int half  = lane >> 4;      // 0 -> K slots 0,1 ; 1 -> K slots 2,3
    const int l15   = lane & 15;
    const int row   = tileM * 16 + l15;   // this lane's A/C row (exact: N % 16 == 0)
    const int col   = tileN * 16 + l15;   // this lane's B/C column

    const float rdeg = 1.0f / fmaxf(deg[row], 1.0f);

    const float* __restrict__ xrow = X   + (size_t)row * K;
    const float* __restrict__ grow = AGG + (size_t)row * K;

    v8f acc = {};

    // self term: X @ Wself
    for (int k = 0; k < K; k += 4) {
        const int kk = k + half * 2;
        v2f a, b;
        a.x = xrow[kk];
        a.y = xrow[kk + 1];
        b.x = Wself[(size_t)kk * Fout + col];
        b.y = Wself[(size_t)(kk + 1) * Fout + col];
        acc = __builtin_amdgcn_wmma_f32_16x16x4_f32(
            /*neg_a=*/false, a, /*neg_b=*/false, b,
            /*c_mod=*/(short)0, acc, /*reuse_a=*/false, /*reuse_b=*/false);
    }
    // neighbor term: (AGG * 1/deg) @ Wneigh
    for (int k = 0; k < K; k += 4) {
        const int kk = k + half * 2;
        v2f a, b;
        a.x = grow[kk] * rdeg;
        a.y = grow[kk + 1] * rdeg;
        b.x = Wneigh[(size_t)kk * Fout + col];
        b.y = Wneigh[(size_t)(kk + 1) * Fout + col];
        acc = __builtin_amdgcn_wmma_f32_16x16x4_f32(
            false, a, false, b, (short)0, acc, false, false);
    }

    const float bv = bias[col];
#pragma unroll
    for (int i = 0; i < 8; ++i) {
        float v = acc[i] + bv;
        if (do_relu) v = fmaxf(v, 0.0f);
        const int r = tileM * 16 + i + half * 8;   // C/D layout: VGPR i -> rows i, i+8
        out[(size_t)r * Fout + col] = v;
    }
}

// ---------------------------------------------------------------- launcher
extern "C" void kernel_launch(void* const* d_in, const int* in_sizes, int n_in,
                              void* d_out, int out_size, void* d_ws, size_t ws_size,
                              hipStream_t stream) {
    const float* in_feat = (const float*)d_in[0];
    const int*   src     = (const int*)  d_in[1];
    const int*   dst     = (const int*)  d_in[2];
    const float* Ws1     = (const float*)d_in[3];
    const float* Wn1     = (const float*)d_in[4];
    const float* b1      = (const float*)d_in[5];
    const float* Ws2     = (const float*)d_in[6];
    const float* Wn2     = (const float*)d_in[7];
    const float* b2      = (const float*)d_in[8];
    float*       out     = (float*)d_out;

    float* ws   = (float*)d_ws;
    float* deg  = ws;                                    // NNODES floats (padded to 128K)
    float* agg1 = ws + (1 << 17);                        // NNODES*128 floats
    float* agg2 = agg1 + (size_t)NNODES * FDIM;          // NNODES*128 floats
    float* h1   = agg2 + (size_t)NNODES * FDIM;          // NNODES*128 floats (no zeroing needed)

    // zero deg + agg1 + agg2 in one pass
    const int zero_floats = (1 << 17) + 2 * NNODES * FDIM;    // 25,731,072 (divisible by 4)
    zero_f4_kernel<<<2048, 256, 0, stream>>>((float4*)ws, zero_floats / 4);

    // in-degree (shared by both layers)
    deg_count_kernel<<<(NEDGES + 255) / 256, 256, 0, stream>>>(dst, deg, NEDGES);

    // ---- layer 1 ----
    scatter_add_kernel<<<NEDGES / 8, 256, 0, stream>>>(in_feat, src, dst, agg1, NEDGES);
    // 6250 M-tiles * 8 N-tiles = 50000 waves, 8 waves/block
    sage_gemm_kernel<<<6250, 256, 0, stream>>>(in_feat, agg1, deg, Ws1, Wn1, b1, h1,
                                               /*K=*/128, /*Fout=*/128, /*ntilesN=*/8,
                                               /*relu=*/1);

    // ---- layer 2 ----
    scatter_add_kernel<<<NEDGES / 8, 256, 0, stream>>>(h1, src, dst, agg2, NEDGES);
    // 6250 M-tiles * 4 N-tiles = 25000 waves
    sage_gemm_kernel<<<3125, 256, 0, stream>>>(h1, agg2, deg, Ws2, Wn2, b2, out,
                                               /*K=*/128, /*Fout=*/64, /*ntilesN=*/4,
                                               /*relu=*/0);
}